// PixelEffectModule_5669356835643
// MI455X (gfx1250) — compile-verified
//
#include <hip/hip_runtime.h>
#include <stdint.h>

// ---------------------------------------------------------------------------
// PixelEffect (oil-paint filter): 8-bin window histogram + argmax-bin mean,
// stride-8 11x11 window, fused 8x nearest upsample. HBM-bound (~103 MB,
// ~0.08 GFLOP) => TDM staging + LDS compute.
// Round 3: iterated TDM re-strides tiles into a padded (stride-72) LDS layout
// so window rows load as b64/b128; u64 packed bin counters; rolled loops to
// keep static code compact (round 2 fully unrolled into 7081 instrs).
// ---------------------------------------------------------------------------

typedef unsigned int       v4u  __attribute__((ext_vector_type(4)));
typedef int                v8i  __attribute__((ext_vector_type(8)));
typedef int                v4i  __attribute__((ext_vector_type(4)));
typedef float              f4   __attribute__((ext_vector_type(4)));
typedef unsigned long long u64;

constexpr int H   = 2048;
constexpr int W   = 2048;
constexpr int NB  = 8;                    // param_num_bins
constexpr int K   = 11;                   // param_kernel_size
constexpr int PS  = 8;                    // param_pixel_size (conv stride, upsample)
constexpr int PAD = (K - 1) / 2;          // 5
constexpr int TILE    = 8;                // 8x8 coarse outputs per block
constexpr int IN_TILE = (TILE - 1) * PS + K;   // 67 staged rows/cols per block
constexpr int SW  = 72;                   // padded LDS row stride (floats / bytes)
constexpr int CHS = IN_TILE * SW;         // per-channel LDS float extent (4824)

__device__ __forceinline__ void store_replicated(
    float vr, float vg, float vb, int oy, int ox, float* __restrict__ out)
{
    const size_t HWsz = (size_t)H * (size_t)W;
    const f4 pr = { vr, vr, vr, vr };
    const f4 pg = { vg, vg, vg, vg };
    const f4 pb = { vb, vb, vb, vb };
    #pragma unroll
    for (int i = 0; i < PS; ++i) {
        const size_t row = ((size_t)(oy * PS + i)) * W + (size_t)ox * PS;
        __builtin_nontemporal_store(pr, (f4*)(out + row));
        __builtin_nontemporal_store(pr, (f4*)(out + row + 4));
        __builtin_nontemporal_store(pg, (f4*)(out + HWsz + row));
        __builtin_nontemporal_store(pg, (f4*)(out + HWsz + row + 4));
        __builtin_nontemporal_store(pb, (f4*)(out + 2 * HWsz + row));
        __builtin_nontemporal_store(pb, (f4*)(out + 2 * HWsz + row + 4));
    }
}

// Argmax over 8 packed u8 counts; strict '>' keeps first index (jnp.argmax).
__device__ __forceinline__ int argmax8(u64 cnt, unsigned& bestCnt)
{
    int bestBin = 0;
    bestCnt = (unsigned)(cnt & 0xffull);
    #pragma unroll
    for (int b = 1; b < NB; ++b) {
        const unsigned c = (unsigned)((cnt >> (8 * b)) & 0xffull);
        if (c > bestCnt) { bestCnt = c; bestBin = b; }
    }
    return bestBin;
}

__global__ __launch_bounds__(32)
void pixel_effect_kernel(const float* __restrict__ rgb, float* __restrict__ out)
{
    // Padded layout: 3 channels x 67 rows x stride 72 floats (57,888 B) plus
    // one bin byte per slot (4,824 B) = 62.7 KB LDS.
    __shared__ __attribute__((aligned(16))) float         smem[3 * CHS];
    __shared__ __attribute__((aligned(16))) unsigned char sbin[CHS];

    const int bx  = blockIdx.x, by = blockIdx.y;
    const int lid = threadIdx.x;

    // Clamping occurs only on left/top edges (last window row/col = 2045).
    const int y0  = by * TILE * PS - PAD;
    const int x0  = bx * TILE * PS - PAD;
    const int cy0 = y0 < 0 ? 0 : y0;
    const int cx0 = x0 < 0 ? 0 : x0;
    const int ry0 = cy0 - y0;                  // 0 unless by==0 (then 5)
    const int rx0 = cx0 - x0;                  // 0 unless bx==0 (then 5)
    const int chh = IN_TILE - ry0;
    const int cw  = IN_TILE - rx0;

    // ---- TDM: per channel, iterated 1-row tiles re-strided into LDS -------
    // D# per CDNA5 ISA ch.8/9.3: tile = cw x 1, iterate chh times with
    // global_addr_increment = W and lds_addr_increment = SW (data units),
    // landing the clamped region at padded offset (ry0*SW + rx0).
    #pragma unroll
    for (int ch = 0; ch < 3; ++ch) {
        const unsigned lds_addr =
            (unsigned)(uintptr_t)(void*)&smem[ch * CHS + ry0 * SW + rx0];
        const u64 ga = (u64)(uintptr_t)
            (rgb + (size_t)ch * H * W + (size_t)cy0 * W + cx0);

        v4u g0 = { 1u,                                       // count=1, user mode
                   lds_addr,                                 // lds_addr [63:32]
                   (unsigned)(ga & 0xffffffffull),           // global_addr lo
                   (unsigned)((ga >> 32) & 0x01ffffffull)    // global_addr hi
                       | 0x80000000u };                      // type=2 ("image")

        v8i g1 = { (int)((2u << 16) | (1u << 19)),           // 4B elems, iterate_enable
                   (int)(((unsigned)W & 0xffffu) << 16),     // tensor_dim0 lo16
                   (int)((((unsigned)W >> 16) & 0xffffu)
                         | (((unsigned)H & 0xffffu) << 16)), // dim0 hi | dim1 lo
                   (int)((((unsigned)H >> 16) & 0xffffu)
                         | ((unsigned)cw << 16)),            // dim1 hi | tile_dim0=cw
                   (int)1,                                   // tile_dim1=1, tile_dim2=0
                   (int)W,                                   // dim0_stride
                   0, 0 };                                   // dim1_stride unused

        v4i g2 = { 0,                  // tensor_dim2 (unused)
                   SW,                 // lds_addr_increment (data units)
                   W,                  // global_addr_increment lo32 (data units)
                   (chh - 1) << 16 };  // incr hi16=0 | iterate_count (0 => 1x)
        v4i g3 = { 0, 0, 0, 0 };

#if defined(__clang_major__) && (__clang_major__ >= 23)
        v8i gpad = { 0, 0, 0, 0, 0, 0, 0, 0 };
        __builtin_amdgcn_tensor_load_to_lds(g0, g1, g2, g3, gpad, 0);
#else
        __builtin_amdgcn_tensor_load_to_lds(g0, g1, g2, g3, 0);
#endif
    }
    __builtin_amdgcn_s_wait_tensorcnt(0);
    asm volatile("" ::: "memory");

    // ---- Precompute bin per padded slot (garbage cols/halo never read) ----
    for (int i = lid; i < CHS; i += 32) {
        const float m = (smem[i] + smem[CHS + i] + smem[2 * CHS + i]) * (1.0f / 3.0f);
        int b = (int)(m * ((float)NB / 256.0f));
        b = b < 0 ? 0 : (b > NB - 1 ? NB - 1 : b);
        sbin[i] = (unsigned char)b;
    }
    asm volatile("" ::: "memory");   // single wave: LDS in-order, no barrier

    if (bx != 0 && by != 0) {
        // ================= interior fast path: full 67x67 valid ============
        #pragma unroll 1
        for (int t = 0; t < 2; ++t) {
            const int o  = lid + t * 32;
            const int ly = o >> 3;
            const int lx = o & 7;

            // Pass 1: row-wise b64+b32 bin loads, u64 packed counters.
            u64 cnt = 0;
            #pragma unroll 1
            for (int ky = 0; ky < K; ++ky) {
                const int base = (ly * PS + ky) * SW + lx * PS;
                const u64      a  = *(const u64*)(sbin + base);       // taps 0..7
                const unsigned b4 = *(const unsigned*)(sbin + base + 8); // taps 8..10
                #pragma unroll
                for (int j = 0; j < 8; ++j)
                    cnt += 1ull << ((((unsigned)(a >> (8 * j)) & 7u) * 8u));
                #pragma unroll
                for (int j = 0; j < 3; ++j)
                    cnt += 1ull << ((((b4 >> (8 * j)) & 7u) * 8u));
            }
            unsigned bestCnt;
            const int bestBin = argmax8(cnt, bestCnt);

            // Pass 2: row-wise b128 float loads, masked FMA accumulation.
            float sr = 0.0f, sg = 0.0f, sb = 0.0f;
            #pragma unroll 1
            for (int ky = 0; ky < K; ++ky) {
                const int base = (ly * PS + ky) * SW + lx * PS;
                const u64      a  = *(const u64*)(sbin + base);
                const unsigned b4 = *(const unsigned*)(sbin + base + 8);
                float fr[12], fg[12], fb[12];
                #pragma unroll
                for (int q = 0; q < 3; ++q) {
                    *(f4*)&fr[4 * q] = *(const f4*)(smem + base + 4 * q);
                    *(f4*)&fg[4 * q] = *(const f4*)(smem + CHS + base + 4 * q);
                    *(f4*)&fb[4 * q] = *(const f4*)(smem + 2 * CHS + base + 4 * q);
                }
                #pragma unroll
                for (int j = 0; j < K; ++j) {
                    const unsigned bj = (j < 8)
                        ? ((unsigned)(a >> (8 * j)) & 0xffu)
                        : ((b4 >> (8 * (j - 8))) & 0xffu);
                    const float m = (bj == (unsigned)bestBin) ? 1.0f : 0.0f;
                    sr = fmaf(m, fr[j], sr);
                    sg = fmaf(m, fg[j], sg);
                    sb = fmaf(m, fb[j], sb);
                }
            }

            const float inv = 1.0f / (float)bestCnt;
            store_replicated(sr * inv, sg * inv, sb * inv,
                             by * TILE + ly, bx * TILE + lx, out);
        }
    } else {
        // ================= edge path (63 blocks): bounds-checked ===========
        const int rx1 = rx0 + cw, ry1 = ry0 + chh;
        #pragma unroll 1
        for (int t = 0; t < 2; ++t) {
            const int o  = lid + t * 32;
            const int ly = o >> 3;
            const int lx = o & 7;

            u64 cnt = 0;
            #pragma unroll 1
            for (int ky = 0; ky < K; ++ky) {
                const int rr = ly * PS + ky;
                if (rr < ry0 || rr >= ry1) continue;      // zero-pad: no bin
                const int rb = rr * SW;
                #pragma unroll 1
                for (int kx = 0; kx < K; ++kx) {
                    const int cc = lx * PS + kx;
                    if (cc < rx0 || cc >= rx1) continue;
                    cnt += 1ull << (((unsigned)sbin[rb + cc] & 7u) * 8u);
                }
            }
            unsigned bestCnt;
            const int bestBin = argmax8(cnt, bestCnt);

            float sr = 0.0f, sg = 0.0f, sb = 0.0f;
            #pragma unroll 1
            for (int ky = 0; ky < K; ++ky) {
                const int rr = ly * PS + ky;
                if (rr < ry0 || rr >= ry1) continue;
                const int rb = rr * SW;
                #pragma unroll 1
                for (int kx = 0; kx < K; ++kx) {
                    const int cc = lx * PS + kx;
                    if (cc < rx0 || cc >= rx1) continue;
                    const int idx = rb + cc;
                    if ((int)sbin[idx] == bestBin) {
                        sr += smem[idx];
                        sg += smem[CHS + idx];
                        sb += smem[2 * CHS + idx];
                    }
                }
            }

            const float inv = 1.0f / (float)bestCnt;
            store_replicated(sr * inv, sg * inv, sb * inv,
                             by * TILE + ly, bx * TILE + lx, out);
        }
    }
}

extern "C" void kernel_launch(void* const* d_in, const int* in_sizes, int n_in,
                              void* d_out, int out_size, void* d_ws, size_t ws_size,
                              hipStream_t stream)
{
    (void)in_sizes; (void)n_in; (void)d_ws; (void)ws_size; (void)out_size;
    const float* rgb = (const float*)d_in[0];   // (1,3,2048,2048) f32
    float* out = (float*)d_out;                 // (1,3,2048,2048) f32

    dim3 grid(W / (TILE * PS), H / (TILE * PS));   // 32 x 32 blocks
    dim3 block(32);                                 // one wave32 per block
    pixel_effect_kernel<<<grid, block, 0, stream>>>(rgb, out);
}